// GATv2_37761352467026
// MI455X (gfx1250) — compile-verified
//
#include <hip/hip_runtime.h>
#include <hip/hip_bf16.h>
#include <math.h>

#define NEG_SLOPE 0.2f

typedef __attribute__((ext_vector_type(2))) float v2f;
typedef __attribute__((ext_vector_type(8))) float v8f;
typedef int v4i_vs __attribute__((vector_size(16)));   // matches builtin param type

__device__ __forceinline__ float lrelu(float v) { return v > 0.f ? v : NEG_SLOPE * v; }

// float atomic max via signed/unsigned int ordering trick (works with -inf init)
__device__ __forceinline__ void atomic_max_f32(float* addr, float val) {
  int* ai = (int*)addr;
  if (val >= 0.f) {
    atomicMax(ai, __float_as_int(val));
  } else {
    atomicMin((unsigned int*)ai, (unsigned int)__float_as_int(val));
  }
}

// src/dst for edge e, with implicit self-loops appended after the E real edges
__device__ __forceinline__ void edge_sd(const int* __restrict__ ei, long long E,
                                        long long e, int& s, int& d) {
  if (e < E) { s = ei[e]; d = ei[E + e]; }
  else       { s = d = (int)(e - E); }
}

__global__ void fill_f32(float* __restrict__ p, long long n, float v) {
  long long i = (long long)blockIdx.x * blockDim.x + threadIdx.x;
  if (i < n) p[i] = v;
}

// ---- CDNA5 async global->LDS staging (ASYNCcnt path), with safe fallback ----
__device__ __forceinline__ void stage_b128(const float* __restrict__ g, float* l) {
#if __has_builtin(__builtin_amdgcn_global_load_async_to_lds_b128)
  __builtin_amdgcn_global_load_async_to_lds_b128(
      (__attribute__((address_space(1))) v4i_vs*)g,
      (__attribute__((address_space(3))) v4i_vs*)l, 0, 0);
#else
  *(float4*)l = *(const float4*)g;   // global_load_b128 + ds_store_b128
#endif
}

__device__ __forceinline__ void stage_wait() {
#if __has_builtin(__builtin_amdgcn_global_load_async_to_lds_b128)
#if __has_builtin(__builtin_amdgcn_s_wait_asynccnt)
  __builtin_amdgcn_s_wait_asynccnt(0);
#else
  asm volatile("s_wait_asynccnt 0x0" ::: "memory");
#endif
#endif
  __syncthreads();
}

// ---------------------------------------------------------------------------
// C[M x OUTC] = A[M x K] @ W[K x OUTC] + bias[OUTC], f32, via V_WMMA_F32_16X16X4_F32
// One wave computes one 16x16 output tile, marching K in steps of 4.
// The block's A row-strip is staged once into LDS (async copy) and the WMMA
// A-operands are fed from LDS; W stays L2-resident (64 KB / 8 KB).
// A 16x4 f32 layout: lanes 0-15 M=0..15 / K={0,1}; lanes 16-31 M=0..15 / K={2,3}
// D 16x16 f32 layout: VGPR r, lanes 0-15 -> (M=r, N=lane); lanes 16-31 -> (M=r+8, N=lane-16)
// ---------------------------------------------------------------------------
__global__ void gemm_bias_wmma(const float* __restrict__ A, const float* __restrict__ W,
                               const float* __restrict__ bias, float* __restrict__ C,
                               int M, int K, int OUTC) {
  extern __shared__ float s_lds[];
  const int ntile  = OUTC >> 4;                       // 16-col tiles per row
  const int wpb    = (int)(blockDim.x >> 5);          // waves per block (8)
  const int wave   = (int)(threadIdx.x >> 5);
  const int lane   = (int)(threadIdx.x & 31);
  const int gtile0 = blockIdx.x * wpb;
  const int tileM0 = gtile0 / ntile;
  const int rowBase = tileM0 << 4;
  const int mtiles  = (wpb + ntile - 1) / ntile;      // distinct row-tiles in this block
  const int rowsStaged = mtiles << 4;

  // cooperative async stage of the A row-strip into LDS (all waves participate)
  const int vecPerRow = K >> 2;
  const int totalVec  = rowsStaged * vecPerRow;
  for (int i = (int)threadIdx.x; i < totalVec; i += (int)blockDim.x) {
    const int r = i / vecPerRow;
    const int c = (i - r * vecPerRow) << 2;
    const int gr = min(rowBase + r, M - 1);
    stage_b128(A + (long long)gr * K + c, s_lds + (size_t)r * K + c);
  }
  stage_wait();

  const int gtile = gtile0 + wave;
  const int tileM = gtile / ntile;
  const int row0  = tileM << 4;
  if (row0 < M) {                                     // wave-uniform: EXEC all-1 for WMMA
    const int col0 = (gtile - tileM * ntile) << 4;
    const int m  = lane & 15;
    const int hi = lane >> 4;
    const float* Ar = s_lds + (size_t)(((tileM - tileM0) << 4) + m) * K;

    v8f acc = (v8f)0.f;
#if __has_builtin(__builtin_amdgcn_wmma_f32_16x16x4_f32)
    for (int k0 = 0; k0 < K; k0 += 4) {
      const int ka = k0 + (hi << 1);
      v2f a, b;
      a.x = Ar[ka];                                   // ds_load (LDS-resident A)
      a.y = Ar[ka + 1];
      b.x = W[(long long)ka * OUTC + col0 + m];
      b.y = W[(long long)(ka + 1) * OUTC + col0 + m];
      // 8 args: (neg_a, A, neg_b, B, c_mod, C, reuse_a, reuse_b)
      acc = __builtin_amdgcn_wmma_f32_16x16x4_f32(false, a, false, b,
                                                  (short)0, acc, false, false);
    }
#else
    for (int r = 0; r < 8; ++r) {
      const int lr = ((tileM - tileM0) << 4) + r + (hi << 3);
      float s = 0.f;
      for (int k = 0; k < K; ++k)
        s += s_lds[(size_t)lr * K + k] * W[(long long)k * OUTC + col0 + m];
      acc[r] = s;
    }
#endif

    const float bv = bias[col0 + m];
#pragma unroll
    for (int r = 0; r < 8; ++r) {
      const int row = row0 + r + (hi << 3);
      if (row < M) C[(long long)row * OUTC + col0 + m] = acc[r] + bv;
    }
  }
}

// ---------------------------------------------------------------------------
// Layer-1 edge kernels: H=8 heads x C=16 channels (feature dim 128)
// ---------------------------------------------------------------------------
__global__ void edge_logits1(const float* __restrict__ xl, const float* __restrict__ xr,
                             const int* __restrict__ ei, const float* __restrict__ att,
                             float* __restrict__ logits, long long E, long long Etot) {
  long long idx = (long long)blockIdx.x * blockDim.x + threadIdx.x;
  long long e = idx >> 3;
  if (e >= Etot) return;
  const int h = (int)(idx & 7);
  int s, d; edge_sd(ei, E, e, s, d);
  const float4* pl = (const float4*)(xl + (long long)s * 128 + h * 16);
  const float4* pr = (const float4*)(xr + (long long)d * 128 + h * 16);
  const float4* pa = (const float4*)(att + h * 16);
  float acc = 0.f;
#pragma unroll
  for (int i = 0; i < 4; ++i) {
    float4 a = pl[i], b = pr[i], w = pa[i];
    acc += w.x * lrelu(a.x + b.x);
    acc += w.y * lrelu(a.y + b.y);
    acc += w.z * lrelu(a.z + b.z);
    acc += w.w * lrelu(a.w + b.w);
  }
  logits[e * 8 + h] = acc;
}

__global__ void seg_max1(const float* __restrict__ logits, const int* __restrict__ ei,
                         float* __restrict__ mx, long long E, long long Etot) {
  long long idx = (long long)blockIdx.x * blockDim.x + threadIdx.x;
  long long e = idx >> 3;
  if (e >= Etot) return;
  const int h = (int)(idx & 7);
  int s, d; edge_sd(ei, E, e, s, d);
  atomic_max_f32(&mx[(long long)d * 8 + h], logits[e * 8 + h]);
}

__global__ void exp_denom1(float* __restrict__ logits, const int* __restrict__ ei,
                           const float* __restrict__ mx, float* __restrict__ den,
                           long long E, long long Etot) {
  long long idx = (long long)blockIdx.x * blockDim.x + threadIdx.x;
  long long e = idx >> 3;
  if (e >= Etot) return;
  const int h = (int)(idx & 7);
  int s, d; edge_sd(ei, E, e, s, d);
  const float p = expf(logits[e * 8 + h] - mx[(long long)d * 8 + h]);
  logits[e * 8 + h] = p;                       // overwrite logits with numerator p
  atomicAdd(&den[(long long)d * 8 + h], p);
}

__global__ void aggregate1(const float* __restrict__ xl, const int* __restrict__ ei,
                           const float* __restrict__ p, const float* __restrict__ den,
                           float* __restrict__ out, long long E, long long Etot) {
  long long idx = (long long)blockIdx.x * blockDim.x + threadIdx.x;
  long long e = idx >> 7;
  if (e >= Etot) return;
  const int f = (int)(idx & 127);
  const int h = f >> 4;
  int s, d; edge_sd(ei, E, e, s, d);
  const float alpha = p[e * 8 + h] / den[(long long)d * 8 + h];
  atomicAdd(&out[(long long)d * 128 + f], alpha * xl[(long long)s * 128 + f]);
}

__global__ void bias_elu(float* __restrict__ h, const float* __restrict__ bias, long long n) {
  long long i = (long long)blockIdx.x * blockDim.x + threadIdx.x;
  if (i >= n) return;
  const float v = h[i] + bias[(int)(i & 127)];
  h[i] = v > 0.f ? v : (expf(v) - 1.f);
}

// ---------------------------------------------------------------------------
// Layer-2 edge kernels: 1 head x 16 channels
// ---------------------------------------------------------------------------
__global__ void edge_logits2(const float* __restrict__ hl, const float* __restrict__ hr,
                             const int* __restrict__ ei, const float* __restrict__ att,
                             float* __restrict__ logits, long long E, long long Etot) {
  long long e = (long long)blockIdx.x * blockDim.x + threadIdx.x;
  if (e >= Etot) return;
  int s, d; edge_sd(ei, E, e, s, d);
  const float4* pl = (const float4*)(hl + (long long)s * 16);
  const float4* pr = (const float4*)(hr + (long long)d * 16);
  const float4* pa = (const float4*)att;
  float acc = 0.f;
#pragma unroll
  for (int i = 0; i < 4; ++i) {
    float4 a = pl[i], b = pr[i], w = pa[i];
    acc += w.x * lrelu(a.x + b.x);
    acc += w.y * lrelu(a.y + b.y);
    acc += w.z * lrelu(a.z + b.z);
    acc += w.w * lrelu(a.w + b.w);
  }
  logits[e] = acc;
}

__global__ void seg_max2(const float* __restrict__ logits, const int* __restrict__ ei,
                         float* __restrict__ mx, long long E, long long Etot) {
  long long e = (long long)blockIdx.x * blockDim.x + threadIdx.x;
  if (e >= Etot) return;
  int s, d; edge_sd(ei, E, e, s, d);
  atomic_max_f32(&mx[d], logits[e]);
}

__global__ void exp_denom2(float* __restrict__ logits, const int* __restrict__ ei,
                           const float* __restrict__ mx, float* __restrict__ den,
                           long long E, long long Etot) {
  long long e = (long long)blockIdx.x * blockDim.x + threadIdx.x;
  if (e >= Etot) return;
  int s, d; edge_sd(ei, E, e, s, d);
  const float p = expf(logits[e] - mx[d]);
  logits[e] = p;
  atomicAdd(&den[d], p);
}

__global__ void aggregate2(const float* __restrict__ hl, const int* __restrict__ ei,
                           const float* __restrict__ p, const float* __restrict__ den,
                           float* __restrict__ out, long long E, long long Etot) {
  long long idx = (long long)blockIdx.x * blockDim.x + threadIdx.x;
  long long e = idx >> 4;
  if (e >= Etot) return;
  const int c = (int)(idx & 15);
  int s, d; edge_sd(ei, E, e, s, d);
  const float alpha = p[e] / den[d];
  atomicAdd(&out[(long long)d * 16 + c], alpha * hl[(long long)s * 16 + c]);
}

__global__ void final_bias(const float* __restrict__ agg, const float* __restrict__ bias,
                           float* __restrict__ out, long long n) {
  long long i = (long long)blockIdx.x * blockDim.x + threadIdx.x;
  if (i >= n) return;
  out[i] = agg[i] + bias[(int)(i & 15)];
}

// ---------------------------------------------------------------------------
extern "C" void kernel_launch(void* const* d_in, const int* in_sizes, int n_in,
                              void* d_out, int out_size, void* d_ws, size_t ws_size,
                              hipStream_t stream) {
  const float* x     = (const float*)d_in[0];
  const int*   ei    = (const int*)d_in[1];
  const float* Wl1   = (const float*)d_in[2];
  const float* bl1   = (const float*)d_in[3];
  const float* Wr1   = (const float*)d_in[4];
  const float* br1   = (const float*)d_in[5];
  const float* att1  = (const float*)d_in[6];
  const float* bias1 = (const float*)d_in[7];
  const float* Wl2   = (const float*)d_in[8];
  const float* bl2   = (const float*)d_in[9];
  const float* Wr2   = (const float*)d_in[10];
  const float* br2   = (const float*)d_in[11];
  const float* att2  = (const float*)d_in[12];
  const float* bias2 = (const float*)d_in[13];

  const int F = 128, HC = 128, NC = 16;
  const long long N    = in_sizes[0] / F;
  const long long E    = in_sizes[1] / 2;
  const long long Etot = E + N;            // + self loops

  // workspace carve-up (floats)
  float* ws = (float*)d_ws;
  size_t o = 0;
  float* xl1  = ws + o; o += (size_t)N * 128;
  float* xr1  = ws + o; o += (size_t)N * 128;
  float* hbuf = ws + o; o += (size_t)N * 128;
  float* p1   = ws + o; o += (size_t)Etot * 8;
  float* m1   = ws + o; o += (size_t)N * 8;
  float* den1 = ws + o; o += (size_t)N * 8;
  float* hl   = ws + o; o += (size_t)N * 16;
  float* hr   = ws + o; o += (size_t)N * 16;
  float* p2   = ws + o; o += (size_t)Etot;
  float* m2   = ws + o; o += (size_t)N;
  float* den2 = ws + o; o += (size_t)N;
  float* agg2 = ws + o; o += (size_t)N * 16;
  (void)ws_size; (void)n_in;

  const int B = 256;
  auto cdiv = [](long long a, long long b) { return (int)((a + b - 1) / b); };
  const float NEG_INF = -INFINITY;

  // init accumulators
  fill_f32<<<cdiv(N * 128, B), B, 0, stream>>>(hbuf, N * 128, 0.f);
  fill_f32<<<cdiv(N * 8,   B), B, 0, stream>>>(m1,   N * 8,   NEG_INF);
  fill_f32<<<cdiv(N * 8,   B), B, 0, stream>>>(den1, N * 8,   0.f);
  fill_f32<<<cdiv(N,       B), B, 0, stream>>>(m2,   N,       NEG_INF);
  fill_f32<<<cdiv(N,       B), B, 0, stream>>>(den2, N,       0.f);
  fill_f32<<<cdiv(N * 16,  B), B, 0, stream>>>(agg2, N * 16,  0.f);

  // ---- layer 1 dense transforms (WMMA f32, async-LDS-staged A) ----
  {
    const int ntile = HC / 16;                             // 8
    const long long waves = ((N + 15) / 16) * ntile;
    const int mtiles = (8 + ntile - 1) / ntile;            // 1
    const size_t shmem = (size_t)(mtiles * 16) * F * sizeof(float);   // 8 KB
    gemm_bias_wmma<<<cdiv(waves, 8), B, shmem, stream>>>(x, Wl1, bl1, xl1, (int)N, F, HC);
    gemm_bias_wmma<<<cdiv(waves, 8), B, shmem, stream>>>(x, Wr1, br1, xr1, (int)N, F, HC);
  }

  // ---- layer 1 edge phase ----
  edge_logits1<<<cdiv(Etot * 8, B), B, 0, stream>>>(xl1, xr1, ei, att1, p1, E, Etot);
  seg_max1   <<<cdiv(Etot * 8, B), B, 0, stream>>>(p1, ei, m1, E, Etot);
  exp_denom1 <<<cdiv(Etot * 8, B), B, 0, stream>>>(p1, ei, m1, den1, E, Etot);
  aggregate1 <<<cdiv(Etot * 128, B), B, 0, stream>>>(xl1, ei, p1, den1, hbuf, E, Etot);
  bias_elu   <<<cdiv(N * 128, B), B, 0, stream>>>(hbuf, bias1, N * 128);

  // ---- layer 2 dense transforms (WMMA f32, async-LDS-staged A) ----
  {
    const int ntile = NC / 16;                             // 1
    const long long waves = ((N + 15) / 16) * ntile;
    const int mtiles = (8 + ntile - 1) / ntile;            // 8
    const size_t shmem = (size_t)(mtiles * 16) * HC * sizeof(float);  // 64 KB
    gemm_bias_wmma<<<cdiv(waves, 8), B, shmem, stream>>>(hbuf, Wl2, bl2, hl, (int)N, HC, NC);
    gemm_bias_wmma<<<cdiv(waves, 8), B, shmem, stream>>>(hbuf, Wr2, br2, hr, (int)N, HC, NC);
  }

  // ---- layer 2 edge phase ----
  edge_logits2<<<cdiv(Etot, B), B, 0, stream>>>(hl, hr, ei, att2, p2, E, Etot);
  seg_max2   <<<cdiv(Etot, B), B, 0, stream>>>(p2, ei, m2, E, Etot);
  exp_denom2 <<<cdiv(Etot, B), B, 0, stream>>>(p2, ei, m2, den2, E, Etot);
  aggregate2 <<<cdiv(Etot * 16, B), B, 0, stream>>>(hl, ei, p2, den2, agg2, E, Etot);

  final_bias <<<cdiv(N * 16, B), B, 0, stream>>>(agg2, bias2, (float*)d_out, N * 16);
}